// PaRIConv_46875273069118
// MI455X (gfx1250) — compile-verified
//
#include <hip/hip_runtime.h>

typedef float v2f __attribute__((ext_vector_type(2)));
typedef float v8f __attribute__((ext_vector_type(8)));

static constexpr int Bq    = 8;
static constexpr int Nq    = 4096;
static constexpr int Kq    = 20;
static constexpr int INq   = 64;
static constexpr int OUTq  = 64;
static constexpr int POSq  = Bq * Nq;          // 32768
static constexpr int SLOTS = POSq * Kq;        // 655360
static constexpr int GROUPS = SLOTS / 16;      // 40960
static constexpr float EPSq = 1e-5f;

// workspace float offsets
static constexpr int WS_S    = 0;     // 4   (sum ppf)
static constexpr int WS_S2   = 4;     // 16  (sum ppf_f*ppf_g)
static constexpr int WS_SUM2 = 20;    // 64
static constexpr int WS_SQ2  = 84;    // 64
static constexpr int WS_SUM3 = 148;   // 64
static constexpr int WS_SQ3  = 212;   // 64
static constexpr int WS_SC1  = 512;   // 32
static constexpr int WS_SH1  = 544;   // 32
static constexpr int WS_SC2  = 576;   // 64
static constexpr int WS_SH2  = 640;   // 64
static constexpr int WS_SC3  = 704;   // 64
static constexpr int WS_SH3  = 768;   // 64
static constexpr int WS_BASIS = 1024;                 // POSq*64 floats
static constexpr int WS_P     = 1024 + POSq * INq;    // SLOTS*64 floats (p, then e in-place)

static __device__ __forceinline__ v8f vzero() {
    v8f z = {0.f, 0.f, 0.f, 0.f, 0.f, 0.f, 0.f, 0.f};
    return z;
}

// ---------------- Pass 1: PPF first/second moments (for analytic BN1 stats) --
__global__ void k_ppf_moments(const float* __restrict__ PPF, float* __restrict__ ws) {
    __shared__ float red[20];
    if (threadIdx.x < 20) red[threadIdx.x] = 0.f;
    __syncthreads();
    float s[4] = {0.f, 0.f, 0.f, 0.f};
    float m[16];
#pragma unroll
    for (int i = 0; i < 16; ++i) m[i] = 0.f;
    const int stride = gridDim.x * blockDim.x;
    for (int i = blockIdx.x * blockDim.x + threadIdx.x; i < SLOTS; i += stride) {
        float4 p4 = reinterpret_cast<const float4*>(PPF)[i];
        float p[4] = {p4.x, p4.y, p4.z, p4.w};
#pragma unroll
        for (int f = 0; f < 4; ++f) {
            s[f] += p[f];
#pragma unroll
            for (int g = 0; g < 4; ++g) m[f * 4 + g] += p[f] * p[g];
        }
    }
#pragma unroll
    for (int f = 0; f < 4; ++f) atomicAdd(&red[f], s[f]);
#pragma unroll
    for (int i = 0; i < 16; ++i) atomicAdd(&red[4 + i], m[i]);
    __syncthreads();
    if (threadIdx.x < 20) atomicAdd(&ws[WS_S + threadIdx.x], red[threadIdx.x]);
}

// ---------------- Pass 2: analytic BN1 scale/shift (h = W1*ppf + b1) --------
__global__ void k_bn1(const float* __restrict__ W1, const float* __restrict__ b1,
                      const float* __restrict__ g1, const float* __restrict__ be1,
                      float* __restrict__ ws) {
    const int c = threadIdx.x;
    if (c >= 32) return;
    const float inv = 1.f / (float)SLOTS;
    float mu[4], w[4];
#pragma unroll
    for (int f = 0; f < 4; ++f) { mu[f] = ws[WS_S + f] * inv; w[f] = W1[c * 4 + f]; }
    float dot = 0.f;
#pragma unroll
    for (int f = 0; f < 4; ++f) dot += w[f] * mu[f];
    const float bb = b1[c];
    const float mean = dot + bb;
    float e2 = 0.f;
#pragma unroll
    for (int f = 0; f < 4; ++f)
#pragma unroll
        for (int g = 0; g < 4; ++g) e2 += w[f] * w[g] * (ws[WS_S2 + f * 4 + g] * inv);
    e2 += 2.f * bb * dot + bb * bb;
    const float var = e2 - mean * mean;
    const float sc = g1[c] * rsqrtf(var + EPSq);
    ws[WS_SC1 + c] = sc;
    ws[WS_SH1 + c] = be1[c] - mean * sc;
}

// ---------------- generic BN finalize from sum/sumsq ------------------------
__global__ void k_bnfin(const float* __restrict__ gma, const float* __restrict__ bta,
                        float* __restrict__ ws, int sumOff, int sqOff,
                        int scOff, int shOff, int nch) {
    const int c = threadIdx.x;
    if (c >= nch) return;
    const float inv = 1.f / (float)SLOTS;
    const float mean = ws[sumOff + c] * inv;
    const float var = ws[sqOff + c] * inv - mean * mean;
    const float sc = gma[c] * rsqrtf(var + EPSq);
    ws[scOff + c] = sc;
    ws[shOff + c] = bta[c] - mean * sc;
}

// ---------------- Pass 3: basis = W_basis @ x via WMMA f32 16x16x4 ----------
// one wave per 16 positions; output layout basis[pos][64] for fast gather
__global__ void __launch_bounds__(256) k_basis(const float* __restrict__ x,
                                               const float* __restrict__ Wb,
                                               float* __restrict__ ws) {
    const int tid = blockIdx.x * blockDim.x + threadIdx.x;
    const int wave = tid >> 5;
    const int lane = tid & 31;
    const int lm = lane & 15;
    const int hi = lane >> 4;
    const int pos0 = wave * 16;
    const int b = pos0 >> 12;
    const int n0 = pos0 & (Nq - 1);
    const float* xb = x + b * INq * Nq;
    float* basis = ws + WS_BASIS;

    v8f acc[4];
#pragma unroll
    for (int t = 0; t < 4; ++t) acc[t] = vzero();

    for (int kk = 0; kk < 16; ++kk) {
        const int k0 = kk * 4 + hi * 2;
        v2f bf;
        bf[0] = xb[k0 * Nq + n0 + lm];
        bf[1] = xb[(k0 + 1) * Nq + n0 + lm];
#pragma unroll
        for (int t = 0; t < 4; ++t) {
            const int row = t * 16 + lm;
            v2f af;
            af[0] = Wb[row * 64 + k0];
            af[1] = Wb[row * 64 + k0 + 1];
            acc[t] = __builtin_amdgcn_wmma_f32_16x16x4_f32(false, af, false, bf,
                                                           (short)0, acc[t], false, false);
        }
    }
    const int pos = pos0 + lm;
#pragma unroll
    for (int t = 0; t < 4; ++t) {
#pragma unroll
        for (int r4 = 0; r4 < 2; ++r4) {
            float4 v;
            v.x = acc[t][r4 * 4 + 0];
            v.y = acc[t][r4 * 4 + 1];
            v.z = acc[t][r4 * 4 + 2];
            v.w = acc[t][r4 * 4 + 3];
            reinterpret_cast<float4*>(basis + pos * 64 + t * 16 + hi * 8)[r4] = v;
        }
    }
}

// ---------------- Pass 4: dk1+BN1+ReLU, dk2 (WMMA), gather, p = dk*sel ------
__global__ void __launch_bounds__(256) k_main(const float* __restrict__ PPF,
                                              const int* __restrict__ edges,
                                              const float* __restrict__ W1,
                                              const float* __restrict__ b1,
                                              const float* __restrict__ W2,
                                              const float* __restrict__ b2,
                                              float* __restrict__ ws) {
    __shared__ float shW1[128], shB1[32], shSc1[32], shSh1[32], shB2[64];
    __shared__ float redS[64], redQ[64];
    const int tx = threadIdx.x;
    if (tx < 128) shW1[tx] = W1[tx];
    if (tx < 32) { shB1[tx] = b1[tx]; shSc1[tx] = ws[WS_SC1 + tx]; shSh1[tx] = ws[WS_SH1 + tx]; }
    if (tx < 64) { shB2[tx] = b2[tx]; redS[tx] = 0.f; redQ[tx] = 0.f; }
    __syncthreads();

    const int lane = tx & 31, lm = lane & 15, hi = lane >> 4;
    const int gwave = (blockIdx.x * blockDim.x + tx) >> 5;
    const int nwaves = (gridDim.x * blockDim.x) >> 5;

    // preload W_dk2 A-fragments (64x32) into registers: 4 M-tiles x 8 K-steps
    float a2[4][8][2];
#pragma unroll
    for (int t = 0; t < 4; ++t)
#pragma unroll
        for (int ks = 0; ks < 8; ++ks) {
            const int row = t * 16 + lm;
            const int k0 = ks * 4 + hi * 2;
            a2[t][ks][0] = W2[row * 32 + k0];
            a2[t][ks][1] = W2[row * 32 + k0 + 1];
        }

    float sA[4][8], sQ[4][8];
#pragma unroll
    for (int t = 0; t < 4; ++t)
#pragma unroll
        for (int r = 0; r < 8; ++r) { sA[t][r] = 0.f; sQ[t][r] = 0.f; }

    const int* col = edges + SLOTS;  // edge_index[1]
    float* pbuf = ws + WS_P;
    const float* basis = ws + WS_BASIS;

    for (int g = gwave; g < GROUPS; g += nwaves) {
        const int slot = g * 16 + lm;
        float4 pf = reinterpret_cast<const float4*>(PPF)[slot];
        float pv[4] = {pf.x, pf.y, pf.z, pf.w};
        const int c = col[slot];

        v8f acc[4];
#pragma unroll
        for (int t = 0; t < 4; ++t) acc[t] = vzero();

#pragma unroll
        for (int ks = 0; ks < 8; ++ks) {
            const int k0 = ks * 4 + hi * 2;
            v2f bf;
#pragma unroll
            for (int j = 0; j < 2; ++j) {
                const int kc = k0 + j;
                float hsum = shB1[kc];
#pragma unroll
                for (int f = 0; f < 4; ++f) hsum += shW1[kc * 4 + f] * pv[f];
                const float hv = shSc1[kc] * hsum + shSh1[kc];
                bf[j] = hv > 0.f ? hv : 0.f;  // ReLU(BN1(h))
            }
#pragma unroll
            for (int t = 0; t < 4; ++t) {
                v2f af; af[0] = a2[t][ks][0]; af[1] = a2[t][ks][1];
                acc[t] = __builtin_amdgcn_wmma_f32_16x16x4_f32(false, af, false, bf,
                                                               (short)0, acc[t], false, false);
            }
        }

        const float* selp = basis + (size_t)c * 64;
        float* outp = pbuf + (size_t)slot * 64;
#pragma unroll
        for (int t = 0; t < 4; ++t) {
#pragma unroll
            for (int r4 = 0; r4 < 2; ++r4) {
                float4 sel = reinterpret_cast<const float4*>(selp + t * 16 + hi * 8)[r4];
                float sv[4] = {sel.x, sel.y, sel.z, sel.w};
                float ov[4];
#pragma unroll
                for (int j = 0; j < 4; ++j) {
                    const int r = r4 * 4 + j;
                    const int ch = t * 16 + hi * 8 + r;
                    const float p = (acc[t][r] + shB2[ch]) * sv[j];
                    ov[j] = p;
                    sA[t][r] += p;
                    sQ[t][r] += p * p;
                }
                float4 o; o.x = ov[0]; o.y = ov[1]; o.z = ov[2]; o.w = ov[3];
                reinterpret_cast<float4*>(outp + t * 16 + hi * 8)[r4] = o;
            }
        }
    }

#pragma unroll
    for (int t = 0; t < 4; ++t)
#pragma unroll
        for (int r = 0; r < 8; ++r) {
            const int ch = t * 16 + hi * 8 + r;
            atomicAdd(&redS[ch], sA[t][r]);
            atomicAdd(&redQ[ch], sQ[t][r]);
        }
    __syncthreads();
    if (tx < 64) {
        atomicAdd(&ws[WS_SUM2 + tx], redS[tx]);
        atomicAdd(&ws[WS_SQ2 + tx], redQ[tx]);
    }
}

// ---------------- Pass 5: edge conv (WMMA, W_edge in LDS), in-place over p --
__global__ void __launch_bounds__(256) k_edge(const float* __restrict__ x,
                                              const float* __restrict__ We,
                                              float* __restrict__ ws) {
    __shared__ float shWe[64 * 128];
    __shared__ float shSc2[64], shSh2[64];
    __shared__ float redS[64], redQ[64];
    const int tx = threadIdx.x;
    for (int i = tx; i < 64 * 128; i += 256) shWe[i] = We[i];
    if (tx < 64) { shSc2[tx] = ws[WS_SC2 + tx]; shSh2[tx] = ws[WS_SH2 + tx];
                   redS[tx] = 0.f; redQ[tx] = 0.f; }
    __syncthreads();

    const int lane = tx & 31, lm = lane & 15, hi = lane >> 4;
    const int gwave = (blockIdx.x * blockDim.x + tx) >> 5;
    const int nwaves = (gridDim.x * blockDim.x) >> 5;

    float sA[4][8], sQ[4][8];
#pragma unroll
    for (int t = 0; t < 4; ++t)
#pragma unroll
        for (int r = 0; r < 8; ++r) { sA[t][r] = 0.f; sQ[t][r] = 0.f; }

    for (int g = gwave; g < GROUPS; g += nwaves) {
        const int slot = g * 16 + lm;
        const int pos = slot / Kq;
        const int b = pos >> 12;
        const int n = pos & (Nq - 1);
        const float* xb = x + b * INq * Nq + n;      // + c*Nq per channel
        float* pb = ws + WS_P + (size_t)slot * 64;

        v8f acc[4];
#pragma unroll
        for (int t = 0; t < 4; ++t) acc[t] = vzero();

#pragma unroll
        for (int ks = 0; ks < 32; ++ks) {
            const int k0 = ks * 4 + hi * 2;
            v2f bf;
            if (ks < 16) {
                // cat rows [0,64): relu(BN2(p)) - pad
#pragma unroll
                for (int j = 0; j < 2; ++j) {
                    const int kc = k0 + j;
                    float fv = shSc2[kc] * pb[kc] + shSh2[kc];
                    fv = fv > 0.f ? fv : 0.f;
                    bf[j] = fv - xb[kc * Nq];
                }
            } else {
                // cat rows [64,128): pad
#pragma unroll
                for (int j = 0; j < 2; ++j) {
                    const int kc = k0 + j - 64;
                    bf[j] = xb[kc * Nq];
                }
            }
#pragma unroll
            for (int t = 0; t < 4; ++t) {
                const int row = t * 16 + lm;
                v2f af;
                af[0] = shWe[row * 128 + k0];
                af[1] = shWe[row * 128 + k0 + 1];
                acc[t] = __builtin_amdgcn_wmma_f32_16x16x4_f32(false, af, false, bf,
                                                               (short)0, acc[t], false, false);
            }
        }

        // store e in place (all reads of this slot's p are done) + stats
#pragma unroll
        for (int t = 0; t < 4; ++t) {
#pragma unroll
            for (int r4 = 0; r4 < 2; ++r4) {
                float ov[4];
#pragma unroll
                for (int j = 0; j < 4; ++j) {
                    const int r = r4 * 4 + j;
                    const float e = acc[t][r];
                    ov[j] = e;
                    sA[t][r] += e;
                    sQ[t][r] += e * e;
                }
                float4 o; o.x = ov[0]; o.y = ov[1]; o.z = ov[2]; o.w = ov[3];
                reinterpret_cast<float4*>(pb + t * 16 + hi * 8)[r4] = o;
            }
        }
    }

#pragma unroll
    for (int t = 0; t < 4; ++t)
#pragma unroll
        for (int r = 0; r < 8; ++r) {
            const int ch = t * 16 + hi * 8 + r;
            atomicAdd(&redS[ch], sA[t][r]);
            atomicAdd(&redQ[ch], sQ[t][r]);
        }
    __syncthreads();
    if (tx < 64) {
        atomicAdd(&ws[WS_SUM3 + tx], redS[tx]);
        atomicAdd(&ws[WS_SQ3 + tx], redQ[tx]);
    }
}

// ---------------- Pass 6: BN3 + LeakyReLU + max over K, transpose to [B,C,N] -
__global__ void __launch_bounds__(256) k_final(float* __restrict__ out,
                                               const float* __restrict__ ws) {
    __shared__ float sh[64 * 65];
    const int tx = threadIdx.x;
    const int c = tx & 63, pq = tx >> 6;
    const int pos0 = blockIdx.x * 64;
    const float sc = ws[WS_SC3 + c];
    const float shf = ws[WS_SH3 + c];
    const float* ebuf = ws + WS_P;

    for (int i = 0; i < 16; ++i) {
        const int pl = i * 4 + pq;
        const float* ep = ebuf + (size_t)(pos0 + pl) * Kq * 64 + c;
        float m = -3.4e38f;
#pragma unroll
        for (int k = 0; k < Kq; ++k) {
            float v = sc * ep[k * 64] + shf;
            v = v > 0.f ? v : 0.2f * v;   // LeakyReLU(0.2)
            m = fmaxf(m, v);
        }
        sh[pl * 65 + c] = m;
    }
    __syncthreads();

    const int c2 = tx >> 2, chunk = tx & 3;
    const int b = pos0 >> 12, n0 = pos0 & (Nq - 1);
    float* op = out + (size_t)(b * 64 + c2) * Nq + n0 + chunk * 16;
#pragma unroll
    for (int j4 = 0; j4 < 4; ++j4) {
        float4 v;
        v.x = sh[(chunk * 16 + j4 * 4 + 0) * 65 + c2];
        v.y = sh[(chunk * 16 + j4 * 4 + 1) * 65 + c2];
        v.z = sh[(chunk * 16 + j4 * 4 + 2) * 65 + c2];
        v.w = sh[(chunk * 16 + j4 * 4 + 3) * 65 + c2];
        reinterpret_cast<float4*>(op)[j4] = v;
    }
}

extern "C" void kernel_launch(void* const* d_in, const int* in_sizes, int n_in,
                              void* d_out, int out_size, void* d_ws, size_t ws_size,
                              hipStream_t stream) {
    (void)in_sizes; (void)n_in; (void)out_size; (void)ws_size;
    const float* x      = (const float*)d_in[0];
    const float* PPF    = (const float*)d_in[1];
    const int*   edge   = (const int*)d_in[2];
    // d_in[3] = bs (compile-time constant 8)
    const float* W_basis = (const float*)d_in[4];
    const float* W_dk1   = (const float*)d_in[5];
    const float* b_dk1   = (const float*)d_in[6];
    const float* g_dk1   = (const float*)d_in[7];
    const float* be_dk1  = (const float*)d_in[8];
    const float* W_dk2   = (const float*)d_in[9];
    const float* b_dk2   = (const float*)d_in[10];
    const float* g_act   = (const float*)d_in[11];
    const float* be_act  = (const float*)d_in[12];
    const float* W_edge  = (const float*)d_in[13];
    const float* g_edge  = (const float*)d_in[14];
    const float* be_edge = (const float*)d_in[15];
    float* out = (float*)d_out;
    float* ws  = (float*)d_ws;

    hipMemsetAsync(ws, 0, 1024 * sizeof(float), stream);   // zero stats region
    k_ppf_moments<<<256, 256, 0, stream>>>(PPF, ws);
    k_bn1<<<1, 32, 0, stream>>>(W_dk1, b_dk1, g_dk1, be_dk1, ws);
    k_basis<<<256, 256, 0, stream>>>(x, W_basis, ws);       // 2048 waves, 16 pos each
    k_main<<<640, 256, 0, stream>>>(PPF, edge, W_dk1, b_dk1, W_dk2, b_dk2, ws);
    k_bnfin<<<1, 64, 0, stream>>>(g_act, be_act, ws, WS_SUM2, WS_SQ2, WS_SC2, WS_SH2, 64);
    k_edge<<<640, 256, 0, stream>>>(x, W_edge, ws);
    k_bnfin<<<1, 64, 0, stream>>>(g_edge, be_edge, ws, WS_SUM3, WS_SQ3, WS_SC3, WS_SH3, 64);
    k_final<<<512, 256, 0, stream>>>(out, ws);
}